// GAT_Net_test_41979010351250
// MI455X (gfx1250) — compile-verified
//
#include <hip/hip_runtime.h>
#include <hip/hip_bf16.h>

#define D_DIM 128
#define H_DIM 2
#define C_DIM 128
#define HC    256
#define NEG_SLOPE 0.2f

typedef float v2f __attribute__((ext_vector_type(2)));
typedef float v8f __attribute__((ext_vector_type(8)));

// ---------------- float atomic max (CAS loop -> global_atomic_cmpswap) ------
__device__ inline void atomicMaxF(float* addr, float val) {
    unsigned int* ua = (unsigned int*)addr;
    unsigned int old = __float_as_uint(*addr);
    while (__uint_as_float(old) < val) {
        unsigned int assumed = old;
        old = atomicCAS(ua, assumed, __float_as_uint(val));
        if (old == assumed) break;
    }
}

// ---------------- init kernels ----------------------------------------------
__global__ void k_init_out(float* out, const float* __restrict__ bias, int total) {
    int t = blockIdx.x * blockDim.x + threadIdx.x;
    if (t < total) out[t] = bias[t & (C_DIM - 1)];
}

__global__ void k_init_node(float* m, float* den, int total) {
    int t = blockIdx.x * blockDim.x + threadIdx.x;
    if (t < total) { m[t] = -3.402823466e38f; den[t] = 0.0f; }
}

__global__ void k_zero1(float* p) { if (threadIdx.x == 0 && blockIdx.x == 0) p[0] = 0.0f; }

// ---------------- sum of edge weights ---------------------------------------
__global__ void k_reduce_ew(const float* __restrict__ ew, float* sumw, int E) {
    __shared__ float sm[256];
    int t = blockIdx.x * 256 + threadIdx.x;
    float v = 0.0f;
    for (int i = t; i < E; i += gridDim.x * 256) v += ew[i];
    sm[threadIdx.x] = v;
    __syncthreads();
    for (int s = 128; s > 0; s >>= 1) {
        if (threadIdx.x < s) sm[threadIdx.x] += sm[threadIdx.x + s];
        __syncthreads();
    }
    if (threadIdx.x == 0) atomicAdd(sumw, sm[0]);
}

// ---------------- per-head scalar s_h = dot(W_e[h], att_edge[h]) ------------
__global__ void k_edge_scalar(const float* __restrict__ W_e,
                              const float* __restrict__ att_edge, float* s) {
    __shared__ float sm[HC];
    int t = threadIdx.x;                  // 256 threads, head = t>>7
    sm[t] = W_e[t] * att_edge[t];
    __syncthreads();
    for (int off = 64; off > 0; off >>= 1) {
        if ((t & 127) < off) sm[t] += sm[t + off];
        __syncthreads();
    }
    if ((t & 127) == 0) s[t >> 7] = sm[t];
}

// ---------------- fp32 WMMA GEMM: xp = x @ W  (N x 128) @ (128 x 256) -------
// one wave computes one 16x16 tile of xp; block = 8 waves = 8 N-tiles
__global__ void k_gemm_wmma(const float* __restrict__ x,
                            const float* __restrict__ W,
                            float* __restrict__ xp, int N) {
    int wave = threadIdx.x >> 5;
    int lane = threadIdx.x & 31;
    int m0 = blockIdx.x * 16;
    int n0 = (blockIdx.y * 8 + wave) * 16;

    int rc    = lane & 15;          // A-row within tile / B,C column within tile
    int khalf = (lane >> 4) * 2;    // 0 for lanes 0-15, 2 for lanes 16-31

    int arow = m0 + rc;
    if (arow >= N) arow = N - 1;    // clamp (N=50000 is 16-aligned, safety only)
    const float* __restrict__ ap = x + (size_t)arow * D_DIM;

    v8f acc = {};
#pragma unroll
    for (int k = 0; k < D_DIM; k += 4) {
        v2f a, b;
        // A 16x4 fp32 layout: v0 = K=khalf, v1 = K=khalf+1 (per lane half)
        a.x = ap[k + khalf];
        a.y = ap[k + khalf + 1];
        // B 4x16 fp32 layout mirrored: rows K=khalf / khalf+1, col = rc
        b.x = W[(size_t)(k + khalf)     * HC + n0 + rc];
        b.y = W[(size_t)(k + khalf + 1) * HC + n0 + rc];
        acc = __builtin_amdgcn_wmma_f32_16x16x4_f32(
            /*neg_a=*/false, a, /*neg_b=*/false, b,
            /*c_mod=*/(short)0, acc, /*reuse_a=*/false, /*reuse_b=*/false);
    }
    // C/D layout: VGPR i -> row = i + 8*(lane>>4), col = lane&15
    int rquad = (lane >> 4) * 8;
#pragma unroll
    for (int i = 0; i < 8; ++i) {
        int r = m0 + rquad + i;
        if (r < N) xp[(size_t)r * HC + n0 + rc] = acc[i];
    }
}

// ---------------- per-(node,head) attention dots ----------------------------
__global__ void k_node_att(const float* __restrict__ xp,
                           const float* __restrict__ att_src,
                           const float* __restrict__ att_dst,
                           float* a_src, float* a_dst, int NH) {
    int t = blockIdx.x * blockDim.x + threadIdx.x;
    if (t >= NH) return;
    int n = t >> 1, h = t & 1;
    const float* __restrict__ v  = xp + (size_t)n * HC + h * C_DIM;
    const float* __restrict__ as = att_src + h * C_DIM;
    const float* __restrict__ ad = att_dst + h * C_DIM;
    float s1 = 0.0f, s2 = 0.0f;
    for (int c = 0; c < C_DIM; ++c) {
        float xv = v[c];
        s1 += xv * as[c];
        s2 += xv * ad[c];
    }
    a_src[t] = s1;
    a_dst[t] = s2;
}

// ---------------- per-edge alpha + segment max ------------------------------
__global__ void k_edge_alpha(const int* __restrict__ src, const int* __restrict__ dst,
                             const float* __restrict__ ew,
                             const float* __restrict__ a_src, const float* __restrict__ a_dst,
                             const float* __restrict__ sumw, const float* __restrict__ s,
                             float* alpha, float* mbuf, int E, int N, float invE) {
    int e = blockIdx.x * blockDim.x + threadIdx.x;
    int T = E + N;
    if (e >= T) return;
    int se, de; float w;
    if (e < E) { se = src[e]; de = dst[e]; w = ew[e]; }
    else       { se = de = e - E;          w = sumw[0] * invE; }
#pragma unroll
    for (int h = 0; h < H_DIM; ++h) {
        float a = a_src[se * 2 + h] + a_dst[de * 2 + h] + w * s[h];
        a = (a > 0.0f) ? a : NEG_SLOPE * a;     // leaky relu
        alpha[(size_t)e * 2 + h] = a;
        atomicMaxF(&mbuf[de * 2 + h], a);
    }
}

// ---------------- per-edge exp + segment sum (ex overwrites alpha) ----------
__global__ void k_edge_exp(const int* __restrict__ dst,
                           float* alpha, const float* __restrict__ mbuf,
                           float* den, int E, int N) {
    int e = blockIdx.x * blockDim.x + threadIdx.x;
    int T = E + N;
    if (e >= T) return;
    int de = (e < E) ? dst[e] : (e - E);
#pragma unroll
    for (int h = 0; h < H_DIM; ++h) {
        float ex = __expf(alpha[(size_t)e * 2 + h] - mbuf[de * 2 + h]);
        alpha[(size_t)e * 2 + h] = ex;
        atomicAdd(&den[de * 2 + h], ex);
    }
}

// ---------------- aggregation: one wave per edge ----------------------------
__global__ void k_edge_aggr(const int* __restrict__ src, const int* __restrict__ dst,
                            const float* __restrict__ ex, const float* __restrict__ den,
                            const float* __restrict__ xp, float* out, int E, int N) {
    int gid  = blockIdx.x * blockDim.x + threadIdx.x;
    int e    = gid >> 5;
    int lane = gid & 31;
    int T = E + N;
    if (e >= T) return;
    int se, de;
    if (e < E) { se = src[e]; de = dst[e]; }
    else       { se = de = e - E; }
    float c0 = 0.5f * ex[(size_t)e * 2]     / den[de * 2];      // mean over H=2
    float c1 = 0.5f * ex[(size_t)e * 2 + 1] / den[de * 2 + 1];
    const float* __restrict__ v0 = xp + (size_t)se * HC;
    const float* __restrict__ v1 = v0 + C_DIM;
    float* o = out + (size_t)de * C_DIM;
#pragma unroll
    for (int i = 0; i < 4; ++i) {
        int c = lane + i * 32;
        atomicAdd(&o[c], c0 * v0[c] + c1 * v1[c]);
    }
}

// ---------------------------------------------------------------------------
extern "C" void kernel_launch(void* const* d_in, const int* in_sizes, int n_in,
                              void* d_out, int out_size, void* d_ws, size_t ws_size,
                              hipStream_t stream) {
    const float* x        = (const float*)d_in[0];
    const int*   ei       = (const int*)  d_in[1];
    const float* ew       = (const float*)d_in[2];
    const float* W        = (const float*)d_in[3];
    const float* W_e      = (const float*)d_in[4];
    const float* att_src  = (const float*)d_in[5];
    const float* att_dst  = (const float*)d_in[6];
    const float* att_edge = (const float*)d_in[7];
    const float* bias     = (const float*)d_in[8];
    float* out = (float*)d_out;

    const int N = in_sizes[0] / D_DIM;
    const int E = in_sizes[1] / 2;
    const int T = E + N;
    const int* src = ei;
    const int* dst = ei + E;

    // workspace layout (floats)
    float* ws = (float*)d_ws;
    size_t off = 0;
    float* xp    = ws + off; off += (size_t)N * HC;        // 12.8M
    float* a_src = ws + off; off += (size_t)N * H_DIM;
    float* a_dst = ws + off; off += (size_t)N * H_DIM;
    float* mbuf  = ws + off; off += (size_t)N * H_DIM;
    float* den   = ws + off; off += (size_t)N * H_DIM;
    float* alpha = ws + off; off += (size_t)T * H_DIM;     // reused as ex
    float* sumw  = ws + off; off += 1;
    float* svec  = ws + off; off += 2;

    const int BS = 256;

    // init
    k_init_out <<<(N * C_DIM + BS - 1) / BS, BS, 0, stream>>>(out, bias, N * C_DIM);
    k_init_node<<<(N * H_DIM + BS - 1) / BS, BS, 0, stream>>>(mbuf, den, N * H_DIM);
    k_zero1    <<<1, 64, 0, stream>>>(sumw);

    // scalar reductions
    k_reduce_ew  <<<256, BS, 0, stream>>>(ew, sumw, E);
    k_edge_scalar<<<1, HC, 0, stream>>>(W_e, att_edge, svec);

    // xp = x @ W via fp32 WMMA
    dim3 ggrid((N + 15) / 16, 2);
    k_gemm_wmma<<<ggrid, BS, 0, stream>>>(x, W, xp, N);

    // node attention dots
    k_node_att<<<(N * H_DIM + BS - 1) / BS, BS, 0, stream>>>(xp, att_src, att_dst,
                                                             a_src, a_dst, N * H_DIM);

    // edge softmax
    float invE = 1.0f / (float)E;
    int eBlocks = (T + BS - 1) / BS;
    k_edge_alpha<<<eBlocks, BS, 0, stream>>>(src, dst, ew, a_src, a_dst, sumw, svec,
                                             alpha, mbuf, E, N, invE);
    k_edge_exp  <<<eBlocks, BS, 0, stream>>>(dst, alpha, mbuf, den, E, N);

    // aggregation, one wave per edge
    long long aggThreads = (long long)T * 32;
    int aBlocks = (int)((aggThreads + BS - 1) / BS);
    k_edge_aggr<<<aBlocks, BS, 0, stream>>>(src, dst, alpha, den, xp, out, E, N);
}